// NMS_1889785610829
// MI455X (gfx1250) — compile-verified
//
#include <hip/hip_runtime.h>
#include <stdint.h>

// ---------------- problem constants ----------------
#define BATCH     16
#define NPRED     25200
#define STRIDE    85        // 4 box + 1 obj + 80 cls
#define NCLS      80
#define KTOP      1024
#define MAX_DET   300
#define CONF_T    0.25f
#define IOU_T     0.45f
#define MAX_WH    7680.0f

#define TILE      192       // preds per phase-1 tile (multiple of 4 -> 16B-aligned spans)
#define T1        192       // threads in phase-1 block (6 waves)
#define T2        1024      // threads in phase-2 block (32 waves)

// float -> order-preserving uint32 (no NaNs in this workload)
__device__ __forceinline__ uint32_t f2sort(float f) {
  uint32_t u = __float_as_uint(f);
  return (u & 0x80000000u) ? ~u : (u | 0x80000000u);
}
__device__ __forceinline__ float sort2f(uint32_t su) {
  uint32_t u = (su & 0x80000000u) ? (su ^ 0x80000000u) : ~su;
  return __uint_as_float(u);
}

// ---------------------------------------------------------------------------
// Kernel 1: stream input via async global->LDS copies (ASYNCcnt path), compute
// per-prediction score = max_c(obj*cls_c), argmax (first-wins, matches jnp),
// pack 64-bit sortable key (sortable(masked_score) << 32) | ~idx.
// ---------------------------------------------------------------------------
__global__ __launch_bounds__(T1)
void score_kernel(const float* __restrict__ x,
                  uint64_t* __restrict__ keybuf,
                  int* __restrict__ clsbuf) {
  __shared__ __align__(16) float tile[TILE * STRIDE];   // 65280 B
  const int img  = blockIdx.y;
  const int t0   = blockIdx.x * TILE;
  const int rows = min(TILE, NPRED - t0);               // 192 or 48 (both %4==0)
  const int tid  = threadIdx.x;

  const float* src = x + ((size_t)img * NPRED + t0) * STRIDE;   // 16B aligned
  const int chunks = (rows * STRIDE) >> 2;                      // 16B chunks
  const uint32_t ldsBase = (uint32_t)(uintptr_t)(void*)tile;    // low 32 bits of flat = LDS offset

  for (int k = tid; k < chunks; k += T1) {
    uint32_t lo = ldsBase + (uint32_t)(k << 4);
    uint32_t go = (uint32_t)(k << 4);
    // GVS mode: global address is 64-bit src plus 32-bit offset go; copies 16B/lane to the LDS address in lo
    asm volatile("global_load_async_to_lds_b128 %0, %1, %2"
                 :: "v"(lo), "v"(go), "s"(src) : "memory");
  }
  asm volatile("s_wait_asynccnt 0" ::: "memory");
  __syncthreads();

  if (tid < rows) {
    const float* r = &tile[tid * STRIDE];   // stride 85 words: conflict-free (gcd(85,64)=1)
    float obj  = r[4];
    float best = obj * r[5];
    int   arg  = 0;
    #pragma unroll 8
    for (int c = 1; c < NCLS; ++c) {
      float p = obj * r[5 + c];
      if (p > best) { best = p; arg = c; }
    }
    float masked = (best > CONF_T) ? best : -1.0f;
    int gi = t0 + tid;
    uint64_t key = ((uint64_t)f2sort(masked) << 32) | (uint32_t)(~(uint32_t)gi);
    keybuf[(size_t)img * NPRED + gi] = key;
    clsbuf[(size_t)img * NPRED + gi] = arg;
  }
}

// ---------------------------------------------------------------------------
// Kernel 2: per-image exact top-1024 of distinct 64-bit keys.
// Byte-wise radix select: 8 passes (vs 64 bitwise), per-wave32 privatized
// 256-bin histograms in LDS (32x256 ints = 32KB) to kill same-bin atomic
// serialization (float scores concentrate in a few exponent bins).
// Then collect exactly 1024 keys, bitonic sort descending in LDS, decode.
// Reproduces jax.lax.top_k stable ordering (keys are distinct).
// ---------------------------------------------------------------------------
__global__ __launch_bounds__(T2)
void topk_kernel(const uint64_t* __restrict__ keybuf,
                 int* __restrict__ topidx,
                 float* __restrict__ topval) {
  __shared__ int      hist[32][256];   // per-wave privatized, 32 KB
  __shared__ int      histSum[256];
  __shared__ uint64_t skeys[KTOP];     // 8 KB
  __shared__ int      selByte, selAcc, cnt;
  const int img = blockIdx.x;
  const int tid = threadIdx.x;
  const int wid = tid >> 5;
  const uint64_t* kk = keybuf + (size_t)img * NPRED;

  // radix select, MSByte -> LSByte; invariant: kneed = rank of target within
  // the subset matching prefix `pre` on bytes above position p.
  uint64_t pre = 0;
  int kneed = KTOP;
  for (int p = 7; p >= 0; --p) {
    for (int i = tid; i < 32 * 256; i += T2) ((int*)hist)[i] = 0;
    __syncthreads();
    const int sh = p * 8;
    const uint64_t maskHi = (p == 7) ? 0ull : (~0ull << (sh + 8));
    for (int i = tid; i < NPRED; i += T2) {
      uint64_t key = kk[i];
      if ((key & maskHi) == pre)
        atomicAdd(&hist[wid][(int)((key >> sh) & 0xFF)], 1);
    }
    __syncthreads();
    if (tid < 256) {
      int s = 0;
      #pragma unroll
      for (int w = 0; w < 32; ++w) s += hist[w][tid];
      histSum[tid] = s;
    }
    __syncthreads();
    if (tid == 0) {
      int acc = 0, b = 255;
      for (; b > 0; --b) {
        int h = histSum[b];
        if (acc + h >= kneed) break;
        acc += h;
      }
      selByte = b;   // bin containing the target rank
      selAcc  = acc; // count strictly above that bin
    }
    __syncthreads();
    pre |= ((uint64_t)(unsigned)selByte) << sh;
    kneed -= selAcc;
    __syncthreads();
  }
  // pre == 1024th-largest key exactly; |{key >= pre}| == KTOP (distinct keys)

  if (tid == 0) cnt = 0;
  __syncthreads();
  for (int i = tid; i < NPRED; i += T2) {
    uint64_t key = kk[i];
    if (key >= pre) {
      int ppos = atomicAdd(&cnt, 1);
      if (ppos < KTOP) skeys[ppos] = key;
    }
  }
  __syncthreads();

  // bitonic sort descending, 1024 elems / 1024 threads
  for (unsigned sz = 2; sz <= KTOP; sz <<= 1) {
    for (unsigned j = sz >> 1; j > 0; j >>= 1) {
      unsigned i = tid;
      unsigned ixj = i ^ j;
      if (ixj > i) {
        uint64_t a = skeys[i], b2 = skeys[ixj];
        bool desc = ((i & sz) == 0);
        if (desc ? (a < b2) : (a > b2)) { skeys[i] = b2; skeys[ixj] = a; }
      }
      __syncthreads();
    }
  }

  {
    uint64_t key = skeys[tid];
    topidx[img * KTOP + tid] = (int)(~(uint32_t)key);
    topval[img * KTOP + tid] = sort2f((uint32_t)(key >> 32));
  }
}

// ---------------------------------------------------------------------------
// Kernel 3: gather boxes, blocked greedy NMS (wave32-sequential within 32-row
// blocks via shuffles, parallel cross-block suppression), compact, emit 300x6.
// ---------------------------------------------------------------------------
__global__ __launch_bounds__(KTOP)
void nms_kernel(const float* __restrict__ x,
                const int* __restrict__ clsbuf,
                const int* __restrict__ topidx,
                const float* __restrict__ topval,
                float* __restrict__ out) {
  __shared__ float ox1[KTOP], oy1[KTOP], ox2[KTOP], oy2[KTOP];
  __shared__ float areaL[KTOP], valL[KTOP];
  __shared__ int   keepL[KTOP], idxL[KTOP];
  __shared__ int   slot[MAX_DET];
  __shared__ int   cntS;

  const int img  = blockIdx.x;
  const int tid  = threadIdx.x;
  const int lane = tid & 31;
  const int wid  = tid >> 5;

  // gather + class-offset boxes (area computed from OFFSET coords, as reference)
  int   pi  = topidx[img * KTOP + tid];
  float val = topval[img * KTOP + tid];
  const float* row = x + ((size_t)img * NPRED + pi) * STRIDE;
  float cx = row[0], cy = row[1], w = row[2], h = row[3];
  float coff = (float)clsbuf[(size_t)img * NPRED + pi] * MAX_WH;
  float x1 = (cx - w * 0.5f) + coff;
  float y1 = (cy - h * 0.5f) + coff;
  float x2 = (cx + w * 0.5f) + coff;
  float y2 = (cy + h * 0.5f) + coff;
  float ar = (x2 - x1) * (y2 - y1);
  int   kj = (val > CONF_T) ? 1 : 0;

  ox1[tid] = x1; oy1[tid] = y1; ox2[tid] = x2; oy2[tid] = y2;
  areaL[tid] = ar; valL[tid] = val; idxL[tid] = pi; keepL[tid] = kj;
  __syncthreads();

  // blocked greedy NMS: 32 blocks of 32 rows -> 64 barriers total
  for (int bi = 0; bi < KTOP / 32; ++bi) {
    if (wid == bi) {
      // wave-sequential within block (wave32, no barriers)
      for (int i = 0; i < 32; ++i) {
        int ki = __shfl(kj, i, 32);
        if (lane > i && ki && kj) {
          int q = bi * 32 + i;
          float xx1 = fmaxf(x1, ox1[q]);
          float yy1 = fmaxf(y1, oy1[q]);
          float xx2 = fminf(x2, ox2[q]);
          float yy2 = fminf(y2, oy2[q]);
          float inter = fmaxf(xx2 - xx1, 0.0f) * fmaxf(yy2 - yy1, 0.0f);
          float uni = areaL[q] + ar - inter;
          if (inter / (uni + 1e-9f) > IOU_T) kj = 0;
        }
      }
      keepL[tid] = kj;   // finalized for this block
    }
    __syncthreads();
    // cross-block suppression: rows in later blocks vs finalized block bi
    if (tid >= (bi + 1) * 32 && kj) {
      int qb = bi * 32;
      for (int i = 0; i < 32; ++i) {
        int q = qb + i;
        if (keepL[q]) {
          float xx1 = fmaxf(x1, ox1[q]);
          float yy1 = fmaxf(y1, oy1[q]);
          float xx2 = fminf(x2, ox2[q]);
          float yy2 = fminf(y2, oy2[q]);
          float inter = fmaxf(xx2 - xx1, 0.0f) * fmaxf(yy2 - yy1, 0.0f);
          float uni = areaL[q] + ar - inter;
          if (inter / (uni + 1e-9f) > IOU_T) { kj = 0; break; }
        }
      }
    }
    __syncthreads();
  }

  // survivors are already in descending-score order (stable) -> serial compact
  if (tid == 0) {
    int c = 0;
    for (int i = 0; i < KTOP && c < MAX_DET; ++i)
      if (keepL[i]) slot[c++] = i;
    cntS = c;
  }
  __syncthreads();

  if (tid < MAX_DET) {
    float o0 = 0.f, o1 = 0.f, o2 = 0.f, o3 = 0.f, o4 = 0.f, o5 = 0.f;
    if (tid < cntS) {
      int j  = slot[tid];
      int p2 = idxL[j];
      const float* r2 = x + ((size_t)img * NPRED + p2) * STRIDE;
      float c0 = r2[0], c1 = r2[1], w2 = r2[2], h2 = r2[3];
      o0 = c0 - w2 * 0.5f; o1 = c1 - h2 * 0.5f;
      o2 = c0 + w2 * 0.5f; o3 = c1 + h2 * 0.5f;
      o4 = valL[j];
      o5 = (float)clsbuf[(size_t)img * NPRED + p2];
    }
    float* op = out + (size_t)img * (MAX_DET * 6) + tid * 6;
    op[0] = o0; op[1] = o1; op[2] = o2; op[3] = o3; op[4] = o4; op[5] = o5;
  }
}

// ---------------------------------------------------------------------------
extern "C" void kernel_launch(void* const* d_in, const int* in_sizes, int n_in,
                              void* d_out, int out_size, void* d_ws, size_t ws_size,
                              hipStream_t stream) {
  (void)in_sizes; (void)n_in; (void)out_size; (void)ws_size;
  const float* x = (const float*)d_in[0];
  float* out = (float*)d_out;

  // workspace layout (~4.97 MB)
  uint64_t* keybuf = (uint64_t*)d_ws;                        // BATCH*NPRED u64
  int*      clsbuf = (int*)(keybuf + (size_t)BATCH * NPRED); // BATCH*NPRED i32
  int*      topidx = clsbuf + (size_t)BATCH * NPRED;         // BATCH*KTOP i32
  float*    topval = (float*)(topidx + (size_t)BATCH * KTOP);// BATCH*KTOP f32

  dim3 g1((NPRED + TILE - 1) / TILE, BATCH);
  score_kernel<<<g1, T1, 0, stream>>>(x, keybuf, clsbuf);
  topk_kernel<<<BATCH, T2, 0, stream>>>(keybuf, topidx, topval);
  nms_kernel<<<BATCH, KTOP, 0, stream>>>(x, clsbuf, topidx, topval, out);
}